// SparseTokenHandler_37185826848774
// MI455X (gfx1250) — compile-verified
//
#include <hip/hip_runtime.h>
#include <hip/hip_bf16.h>
#include <stdint.h>

// Problem constants from the reference: x is [B=8, L=4096, C=1024] f32,
// SPARSE_RATIO = 0.5 -> keep K = 2048 tokens per batch.
#define B_DIM 8
#define L_DIM 4096
#define C_DIM 1024
#define K_KEEP 2048

// Native clang vector type (HIP's float4 class is rejected by
// __builtin_nontemporal_store).
typedef float vfloat4 __attribute__((ext_vector_type(4)));

// ---------------------------------------------------------------------------
// Pass 1: squared L2 norm per token row. One wave32 per row of 1024 floats.
// Each lane: 8x float4 loads (coalesced b128 traffic, default RT policy so x
// becomes L2-resident for pass 3), FMA chain, 5-step xor-shuffle reduction.
// ---------------------------------------------------------------------------
__global__ __launch_bounds__(256) void stk_norms_kernel(
    const float* __restrict__ x, float* __restrict__ norms) {
  const int lane = threadIdx.x & 31;
  const int wave = threadIdx.x >> 5;
  const int row  = blockIdx.x * (blockDim.x >> 5) + wave;
  if (row >= B_DIM * L_DIM) return;

  const vfloat4* __restrict__ src =
      (const vfloat4*)(x + (size_t)row * C_DIM);
  float acc = 0.0f;
#pragma unroll
  for (int i = 0; i < 8; ++i) {
    vfloat4 v = src[i * 32 + lane];
    acc = fmaf(v.x, v.x, acc);
    acc = fmaf(v.y, v.y, acc);
    acc = fmaf(v.z, v.z, acc);
    acc = fmaf(v.w, v.w, acc);
  }
#pragma unroll
  for (int off = 16; off > 0; off >>= 1)
    acc += __shfl_xor(acc, off, 32);
  if (lane == 0) norms[row] = acc;
}

// Key: (bits(norm^2) << 32) | (0xFFFFFFFF - index).
// norm^2 >= 0 so its IEEE bits are monotone as uint32; the inverted index
// makes keys unique and reproduces top_k's lower-index-wins tie-break under
// a descending sort.
__device__ __forceinline__ unsigned long long stk_make_key(float n2, int idx) {
  return ((unsigned long long)__float_as_uint(n2) << 32) |
         (unsigned)(0xFFFFFFFFu - (unsigned)idx);
}

// ---------------------------------------------------------------------------
// Pass 2: per-batch rank selection. One 1024-thread block per batch sorts the
// 4096 keys descending with an in-LDS bitonic sort (32 KB of the 320 KB/WGP
// LDS) and emits the K-th largest key as the selection threshold.
// ---------------------------------------------------------------------------
__global__ __launch_bounds__(1024) void stk_select_kernel(
    const float* __restrict__ norms, unsigned long long* __restrict__ kth) {
  __shared__ unsigned long long keys[L_DIM];
  const int b   = blockIdx.x;
  const int tid = threadIdx.x;

  for (int i = tid; i < L_DIM; i += blockDim.x)
    keys[i] = stk_make_key(norms[b * L_DIM + i], i);
  __syncthreads();

  for (int k = 2; k <= L_DIM; k <<= 1) {
    for (int j = k >> 1; j > 0; j >>= 1) {
      for (int i = tid; i < L_DIM; i += blockDim.x) {
        const int ixj = i ^ j;
        if (ixj > i) {
          const unsigned long long a = keys[i];
          const unsigned long long c = keys[ixj];
          const bool desc = ((i & k) == 0);  // overall descending sort
          if (desc ? (a < c) : (a > c)) {
            keys[i]   = c;
            keys[ixj] = a;
          }
        }
      }
      __syncthreads();
    }
  }
  if (tid == 0) kth[b] = keys[K_KEEP - 1];
}

// ---------------------------------------------------------------------------
// Pass 3: masked copy. One wave32 per token row (branch is wave-uniform).
// Selected rows stream through the gfx1250 async global<->LDS path:
//   global_load_async_to_lds_b128 (th:LU, last use of x -> free L2 lines)
//   -> s_wait_asynccnt 0 ->
//   global_store_async_from_lds_b128 (th:NT, output never re-read)
//   -> s_wait_asynccnt 0
// using a private 4 KB LDS slice per wave (8 waves -> 32 KB/block).
// Dropped rows get non-temporal vectorized zero stores.
// Every output element is written exactly once.
// ---------------------------------------------------------------------------
__global__ __launch_bounds__(256) void stk_scatter_kernel(
    const float* __restrict__ x, const float* __restrict__ norms,
    const unsigned long long* __restrict__ kth, float* __restrict__ out) {
  __shared__ __align__(16) unsigned char stage[8 * 4096];  // 8 waves * 4KB

  const int lane          = threadIdx.x & 31;
  const int wave          = threadIdx.x >> 5;
  const int wavesPerBlock = blockDim.x >> 5;
  const int globalWave    = blockIdx.x * wavesPerBlock + wave;
  const int totalWaves    = gridDim.x * wavesPerBlock;

  // Low 32 bits of a generic pointer to LDS == byte offset within LDS.
  const unsigned ldsBase =
      (unsigned)(uintptr_t)(void*)&stage[wave * 4096];

  for (int row = globalWave; row < B_DIM * L_DIM; row += totalWaves) {
    const int b = row >> 12;    // row / L_DIM
    const int l = row & (L_DIM - 1);

    const unsigned long long key = stk_make_key(norms[row], l);
    const bool sel = (key >= kth[b]);  // wave-uniform

    const float* __restrict__ src = x   + (size_t)row * C_DIM;
    float* __restrict__       dst = out + (size_t)row * C_DIM;

    if (sel) {
#pragma unroll
      for (int i = 0; i < 8; ++i) {
        const unsigned loff = ldsBase + (unsigned)((i * 32 + lane) * 16);
        const unsigned long long ga =
            (unsigned long long)(uintptr_t)(src + (i * 32 + lane) * 4);
        asm volatile("global_load_async_to_lds_b128 %0, %1, off th:TH_LOAD_LU"
                     :: "v"(loff), "v"(ga) : "memory");
      }
      asm volatile("s_wait_asynccnt 0x0" ::: "memory");
#pragma unroll
      for (int i = 0; i < 8; ++i) {
        const unsigned loff = ldsBase + (unsigned)((i * 32 + lane) * 16);
        const unsigned long long ga =
            (unsigned long long)(uintptr_t)(dst + (i * 32 + lane) * 4);
        asm volatile("global_store_async_from_lds_b128 %0, %1, off th:TH_STORE_NT"
                     :: "v"(ga), "v"(loff) : "memory");
      }
      asm volatile("s_wait_asynccnt 0x0" ::: "memory");
    } else {
      const vfloat4 z = {0.0f, 0.0f, 0.0f, 0.0f};
      vfloat4* __restrict__ dst4 = (vfloat4*)dst;
#pragma unroll
      for (int i = 0; i < 8; ++i)
        __builtin_nontemporal_store(z, dst4 + i * 32 + lane);
    }
  }
}

extern "C" void kernel_launch(void* const* d_in, const int* in_sizes, int n_in,
                              void* d_out, int out_size, void* d_ws, size_t ws_size,
                              hipStream_t stream) {
  (void)in_sizes; (void)n_in; (void)out_size; (void)ws_size;

  const float* x   = (const float*)d_in[0];
  float*       out = (float*)d_out;

  // Workspace layout: [B*L floats norms][B uint64 thresholds]
  float* norms = (float*)d_ws;
  unsigned long long* kth =
      (unsigned long long*)((char*)d_ws +
                            (size_t)B_DIM * L_DIM * sizeof(float));

  const int rows = B_DIM * L_DIM;            // 32768
  const int wavesPerBlock = 256 / 32;        // 8

  // Pass 1: norms (one wave per row).
  stk_norms_kernel<<<rows / wavesPerBlock, 256, 0, stream>>>(x, norms);

  // Pass 2: per-batch threshold key via in-LDS bitonic sort.
  stk_select_kernel<<<B_DIM, 1024, 0, stream>>>(norms, kth);

  // Pass 3: masked scatter-copy (one wave per row, async LDS staging,
  // last-use loads + non-temporal stores to protect L2 residency of x).
  stk_scatter_kernel<<<rows / wavesPerBlock, 256, 0, stream>>>(x, norms, kth, out);
}